// Conv_84018150245195
// MI455X (gfx1250) — compile-verified
//
#include <hip/hip_runtime.h>

// Problem constants (match the reference)
#define NN 50000      // nodes
#define NE 800000     // edges
#define DD 64         // feature dim
#define DH 256        // 4*D pooled dim
#define DC 320        // 5*D concat dim
#define DO 64         // output dim

typedef __attribute__((ext_vector_type(2))) float v2f;
typedef __attribute__((ext_vector_type(8))) float v8f;

// ---------------------------------------------------------------------------
// Workspace layout (float units):
//   h        : [0,                NN*DH)            pooled features, N x 256
//   aggSum   : [NN*DH + 0*NN*64,  ...)
//   aggMean  : [NN*DH + 1*NN*64,  ...)   (unnormalized sum)
//   aggMax   : [NN*DH + 2*NN*64,  ...)   (init -inf)
//   m1       : [NN*DH + 3*NN*64,  ...)
//   m2       : [NN*DH + 4*NN*64,  ...)
//   deg      : [NN*DH + 5*NN*64,  +NN)
// ---------------------------------------------------------------------------

__global__ void init_kernel(float* __restrict__ agg, float* __restrict__ deg) {
    long long i = (long long)blockIdx.x * blockDim.x + threadIdx.x;
    const long long NF = (long long)NN * 64;
    if (i < 5 * NF) {
        // region 2 is the max-aggregation buffer: identity of segment_max is -inf
        agg[i] = (i >= 2 * NF && i < 3 * NF) ? -__builtin_inff() : 0.0f;
    } else if (i < 5 * NF + NN) {
        deg[i - 5 * NF] = 0.0f;
    }
}

// ---------------------------------------------------------------------------
// GEMM 1: h(N x 256) = feat(N x 64) @ W_pool(256 x 64)^T + b_pool
// One wave computes one 16x16 tile via 16 chained V_WMMA_F32_16X16X4_F32.
// ---------------------------------------------------------------------------
__global__ void gemm_pool_kernel(const float* __restrict__ feat,
                                 const float* __restrict__ Wp,   // 256 x 64 row-major
                                 const float* __restrict__ bp,   // 256
                                 float* __restrict__ h) {
    const int tiles = (NN / 16) * (DH / 16);              // 3125 * 16 = 50000
    int wave = blockIdx.x * (blockDim.x >> 5) + (threadIdx.x >> 5);
    if (wave >= tiles) return;                            // wave-uniform: EXEC stays full

    int rt = wave / (DH / 16);
    int ct = wave % (DH / 16);
    int row0 = rt * 16, col0 = ct * 16;

    int lane = threadIdx.x & 31;
    int m    = lane & 15;            // A-row within tile / B,C,D column within tile
    int koff = (lane >> 4) << 1;     // lanes 16..31 hold K+2,K+3 (A and B)
    int col  = col0 + m;             // output column (N index)

    float bias = bp[col];
    v8f acc;
#pragma unroll
    for (int i = 0; i < 8; ++i) acc[i] = bias;

    const float* arow = feat + (long long)(row0 + m) * DD;
    const float* bcol = Wp   + (long long)col * DD;       // B[k][n] = Wp[n*64 + k]

#pragma unroll
    for (int kb = 0; kb < DD / 4; ++kb) {                 // 16 K-blocks of 4
        int k = kb * 4;
        v2f a = *(const v2f*)(arow + k + koff);           // A[m][k+koff], A[m][k+koff+1]
        v2f b = *(const v2f*)(bcol + k + koff);           // B[k+koff][col], B[k+koff+1][col]
        acc = __builtin_amdgcn_wmma_f32_16x16x4_f32(
                  false, a, false, b, (short)0, acc, false, false);
    }

    int rbase = row0 + ((lane >> 4) << 3);                // C/D: lanes 16..31 -> rows +8
#pragma unroll
    for (int i = 0; i < 8; ++i)
        h[(long long)(rbase + i) * DH + col] = acc[i];
}

// ---------------------------------------------------------------------------
// Edge scatter: 64 threads per edge (one feature each), fp32 atomics into L2.
// Float-max via the sign-aware int/uint ordering trick (handles -inf init, -0).
// ---------------------------------------------------------------------------
__device__ __forceinline__ void atomicMaxFloat(float* addr, float val) {
    unsigned int u = __float_as_uint(val);
    if (u >> 31) {
        // negative (incl. -0): larger float == smaller uint pattern
        atomicMin((unsigned int*)addr, u);
    } else {
        // non-negative: int ordering == float ordering, beats any negative pattern
        atomicMax((int*)addr, (int)u);
    }
}

__global__ void edge_kernel(const float* __restrict__ h,
                            const float* __restrict__ weight,
                            const int*   __restrict__ src,
                            const int*   __restrict__ dst,
                            float* __restrict__ aggSum,
                            float* __restrict__ aggMean,
                            float* __restrict__ aggMax,
                            float* __restrict__ m1,
                            float* __restrict__ m2,
                            float* __restrict__ deg) {
    int j = threadIdx.x & 63;
    int e = blockIdx.x * (blockDim.x >> 6) + (threadIdx.x >> 6);
    if (e >= NE) return;

    int   s = src[e];
    int   d = dst[e];
    float w = weight[e];

    const float* hr = h + (long long)s * DH;
    float vs = hr[j]        * w;       // sum slice
    float vm = hr[64  + j]  * w;       // mean slice (normalized later)
    float vx = hr[128 + j]  * w;       // max slice
    float hd = hr[192 + j];            // std slice

    long long o = (long long)d * 64 + j;
    atomicAdd(aggSum  + o, vs);
    atomicAdd(aggMean + o, vm);
    atomicMaxFloat(aggMax + o, vx);
    atomicAdd(m1 + o, hd * w);
    atomicAdd(m2 + o, hd * hd * w);
    if (j == 0) atomicAdd(deg + d, 1.0f);
}

// ---------------------------------------------------------------------------
// GEMM 2: out(N x 64) = [feat | h_neigh](N x 320) @ W_neigh(64 x 320)^T + b
// A-fragments of the concatenated matrix are synthesized on the fly:
// normalization, m2 - m1^2, and the deg==0 mask are fused into the fetch.
// K-region (k>>6) is wave-uniform per K-block -> no divergence around WMMA.
// ---------------------------------------------------------------------------
__global__ void gemm_out_kernel(const float* __restrict__ feat,
                                const float* __restrict__ aggSum,
                                const float* __restrict__ aggMean,
                                const float* __restrict__ aggMax,
                                const float* __restrict__ m1,
                                const float* __restrict__ m2,
                                const float* __restrict__ deg,
                                const float* __restrict__ Wn,   // 64 x 320 row-major
                                const float* __restrict__ bn,   // 64
                                float* __restrict__ out) {
    const int tiles = (NN / 16) * (DO / 16);              // 3125 * 4 = 12500
    int wave = blockIdx.x * (blockDim.x >> 5) + (threadIdx.x >> 5);
    if (wave >= tiles) return;

    int rt = wave / (DO / 16);
    int ct = wave % (DO / 16);
    int row0 = rt * 16, col0 = ct * 16;

    int lane = threadIdx.x & 31;
    int m    = lane & 15;
    int koff = (lane >> 4) << 1;
    int col  = col0 + m;
    int r    = row0 + m;                                  // A row for this lane

    float dg  = deg[r];
    float zf  = dg > 0.0f ? 1.0f : 0.0f;                  // where(deg==0 -> 0)
    float inv = 1.0f / fmaxf(dg, 1.0f);                   // 1/safe_deg

    float bias = bn[col];
    v8f acc;
#pragma unroll
    for (int i = 0; i < 8; ++i) acc[i] = bias;

    long long rb   = (long long)r * 64;
    const float* bcol = Wn + (long long)col * DC;         // B[k][n] = Wn[n*320 + k]

#pragma unroll
    for (int kb = 0; kb < DC / 4; ++kb) {                 // 80 K-blocks of 4
        int k      = kb * 4;
        int region = k >> 6;                              // uniform across the wave
        int jj     = (k & 63) + koff;
        long long ro = rb + jj;

        v2f a;
        if (region == 0) {                                // raw features
            a = *(const v2f*)(feat + ro);
        } else if (region == 1) {                         // weighted sum
            a.x = aggSum[ro]     * zf;
            a.y = aggSum[ro + 1] * zf;
        } else if (region == 2) {                         // mean = sum / safe_deg
            a.x = aggMean[ro]     * inv * zf;
            a.y = aggMean[ro + 1] * inv * zf;
        } else if (region == 3) {                         // max (mask the -inf identity)
            a.x = (zf > 0.0f) ? aggMax[ro]     : 0.0f;
            a.y = (zf > 0.0f) ? aggMax[ro + 1] : 0.0f;
        } else {                                          // std = m2/sd - (m1/sd)^2
            float s0 = m1[ro]     * inv;
            float s1 = m1[ro + 1] * inv;
            a.x = (m2[ro]     * inv - s0 * s0) * zf;
            a.y = (m2[ro + 1] * inv - s1 * s1) * zf;
        }

        v2f b = *(const v2f*)(bcol + k + koff);
        acc = __builtin_amdgcn_wmma_f32_16x16x4_f32(
                  false, a, false, b, (short)0, acc, false, false);
    }

    int rbase = row0 + ((lane >> 4) << 3);
#pragma unroll
    for (int i = 0; i < 8; ++i)
        out[(long long)(rbase + i) * DO + col] = acc[i];
}

// ---------------------------------------------------------------------------
extern "C" void kernel_launch(void* const* d_in, const int* in_sizes, int n_in,
                              void* d_out, int out_size, void* d_ws, size_t ws_size,
                              hipStream_t stream) {
    (void)in_sizes; (void)n_in; (void)out_size; (void)ws_size;

    const float* feat   = (const float*)d_in[0];
    const float* weight = (const float*)d_in[1];
    const int*   src    = (const int*)  d_in[2];
    const int*   dst    = (const int*)  d_in[3];
    const float* Wp     = (const float*)d_in[4];
    const float* bp     = (const float*)d_in[5];
    const float* Wn     = (const float*)d_in[6];
    const float* bn     = (const float*)d_in[7];
    float* out = (float*)d_out;

    float* ws      = (float*)d_ws;
    float* h       = ws;                                   // NN*256
    float* agg     = ws + (size_t)NN * DH;
    float* aggSum  = agg + 0 * (size_t)NN * 64;
    float* aggMean = agg + 1 * (size_t)NN * 64;
    float* aggMax  = agg + 2 * (size_t)NN * 64;
    float* m1      = agg + 3 * (size_t)NN * 64;
    float* m2      = agg + 4 * (size_t)NN * 64;
    float* deg     = agg + 5 * (size_t)NN * 64;

    // 1) init aggregation buffers (sums -> 0, max -> -inf, deg -> 0)
    long long initN = 5LL * NN * 64 + NN;
    init_kernel<<<(unsigned)((initN + 255) / 256), 256, 0, stream>>>(agg, deg);

    // 2) pooled projection: 50000 tiles, 8 waves / 256-thread block
    gemm_pool_kernel<<<(50000 + 7) / 8, 256, 0, stream>>>(feat, Wp, bp, h);

    // 3) edge scatter: 4 edges per 256-thread block
    edge_kernel<<<(NE + 3) / 4, 256, 0, stream>>>(h, weight, src, dst,
                                                  aggSum, aggMean, aggMax,
                                                  m1, m2, deg);

    // 4) fused normalize + concat + output projection: 12500 tiles
    gemm_out_kernel<<<(12500 + 7) / 8, 256, 0, stream>>>(feat, aggSum, aggMean,
                                                         aggMax, m1, m2, deg,
                                                         Wn, bn, out);
}